// CustomLoss_37821482008892
// MI455X (gfx1250) — compile-verified
//
#include <hip/hip_runtime.h>
#include <math.h>

// CDNA5 wave32 WMMA f32 16x16x4 fragments:
//   A (16x4):  lanes 0-15 hold M=lane, VGPR0=K0, VGPR1=K1; lanes 16-31 hold M=lane-16, K2/K3
//   B (4x16):  lanes 0-15 hold N=lane, VGPR0=K0, VGPR1=K1; lanes 16-31 hold N=lane-16, K2/K3
//   C (16x16): VGPR r: lanes 0-15 -> (M=r, N=lane); lanes 16-31 -> (M=r+8, N=lane-16)
// K-slot -> data-index mapping within one WMMA is arbitrary (sum over K commutes),
// which we exploit in K3 to turn all fragment loads into contiguous b128 loads.
// Hot kernels are templated on a compile-time D so row offsets become 24-bit
// instruction immediates (no per-step 64-bit address math).
typedef __attribute__((ext_vector_type(2))) float v2f;
typedef __attribute__((ext_vector_type(4))) float v4f;
typedef __attribute__((ext_vector_type(8))) float v8f;

#define LOG2PI 1.8378770664093453f

// ------------------------------------------- K0: counts + packed target bits
__global__ void k_counts(const long long* __restrict__ target,
                         const float* __restrict__ logdet,
                         float* __restrict__ hdr,
                         unsigned* __restrict__ tbits, int B) {
  if (threadIdx.x == 0 && blockIdx.x == 0) {
    float c0 = 0.f, c1 = 0.f, l0 = 0.f, l1 = 0.f;
    for (int w = 0; w < B / 32; ++w) tbits[w] = 0u;
    for (int b = 0; b < B; ++b) {
      if (target[b] == 0) { c0 += 1.f; l0 += logdet[b]; }
      else                { c1 += 1.f; l1 += logdet[b];
                            tbits[b >> 5] |= (1u << (b & 31)); }
    }
    hdr[0] = c0; hdr[1] = c1; hdr[2] = l0; hdr[3] = l1;
  }
}

// ---------------------------- K1: group column sums of mean/log_sd via WMMA
// sum_x[j][d] = sum_b (target[b]==j) * x[b][d]   computed as  mask(2xB) @ x(BxD)
template <int DC>
__global__ void k_group_sums(const float* __restrict__ mean,
                             const float* __restrict__ lsd,
                             const unsigned* __restrict__ tbits,
                             float* __restrict__ sum_mean,   // [2][D]
                             float* __restrict__ sum_lsd,    // [2][D]
                             int B, int Drt) {
  const int D     = DC ? DC : Drt;
  const int lane  = threadIdx.x & 31;
  const int wave  = threadIdx.x >> 5;
  const int dbase = (blockIdx.x * 8 + wave) * 16;   // 16 d-columns per wave
  if (dbase >= D) return;                           // wave-uniform
  const int m     = lane & 15;                      // M row / N col within half
  const int khalf = lane >> 4;                      // 0 -> K{0,1}, 1 -> K{2,3}

  v8f acc_m = {};
  v8f acc_l = {};
  const int d = dbase + m;                          // B-fragment column (=N)
  // per-lane base pointers; row offsets below are compile-time immediates
  const float* pm = mean + (size_t)(2 * khalf) * D + d;
  const float* pl = lsd  + (size_t)(2 * khalf) * D + d;
  for (int w = 0; w < B / 32; ++w) {
    const unsigned word = tbits[w];                 // uniform, L2-hot
    const float* pmw = pm + (size_t)w * 32 * D;     // one bump per 32 rows
    const float* plw = pl + (size_t)w * 32 * D;
#pragma unroll
    for (int j = 0; j < 8; ++j) {
      const int sh0 = j * 4 + 2 * khalf;
      const int t0  = (int)((word >> sh0) & 1u);
      const int t1  = (int)((word >> (sh0 + 1)) & 1u);
      // A: one-hot class mask, rows M=0,1 live, rest zero (no memory traffic)
      v2f a = { (m < 2 && t0 == m) ? 1.0f : 0.0f,
                (m < 2 && t1 == m) ? 1.0f : 0.0f };
      // B: tiles of mean/log_sd, coalesced across lanes (consecutive d per row)
      v2f bm = { pmw[(size_t)(j * 4) * D], pmw[(size_t)(j * 4 + 1) * D] };
      v2f bl = { plw[(size_t)(j * 4) * D], plw[(size_t)(j * 4 + 1) * D] };
      acc_m = __builtin_amdgcn_wmma_f32_16x16x4_f32(false, a, false, bm, (short)0, acc_m, false, false);
      acc_l = __builtin_amdgcn_wmma_f32_16x16x4_f32(false, a, false, bl, (short)0, acc_l, false, false);
    }
  }
  // rows M=0 (acc[0]) and M=1 (acc[1]) on lanes 0-15 hold the group sums
  if (lane < 16) {
    const int dd = dbase + lane;
    sum_mean[dd]     = acc_m[0];
    sum_mean[D + dd] = acc_m[1];
    sum_lsd [dd]     = acc_l[0];
    sum_lsd [D + dd] = acc_l[1];
  }
}

// ------------- K2: per-d finalize + deterministic block partials of Sum lsd,
//               Sum mu^2*inv  (written per block, reduced serially in K4)
__global__ void k_finalize(const float* __restrict__ sum_mean,
                           const float* __restrict__ sum_lsd,
                           const float* __restrict__ hdr,
                           float* __restrict__ inv,     // [2][D]
                           float* __restrict__ muinv,   // [2][D]
                           float* __restrict__ blkpart, // [nblk][4]
                           int D) {
  const int t = threadIdx.x;
  const int d = blockIdx.x * 256 + t;
  const float c0 = hdr[0], c1 = hdr[1];
  const float mu0 = sum_mean[d] / c0, mu1 = sum_mean[D + d] / c1;
  const float l0  = sum_lsd [d] / c0, l1  = sum_lsd [D + d] / c1;
  const float i0  = expf(-2.f * l0),  i1  = expf(-2.f * l1);
  inv[d]       = i0;        inv[D + d]   = i1;
  muinv[d]     = mu0 * i0;  muinv[D + d] = mu1 * i1;

  __shared__ float sh[256][4];
  sh[t][0] = l0; sh[t][1] = l1;
  sh[t][2] = mu0 * mu0 * i0; sh[t][3] = mu1 * mu1 * i1;
  __syncthreads();
  for (int off = 128; off; off >>= 1) {
    if (t < off) { for (int q = 0; q < 4; ++q) sh[t][q] += sh[t + off][q]; }
    __syncthreads();
  }
  if (t == 0) { for (int q = 0; q < 4; ++q) blkpart[blockIdx.x * 4 + q] = sh[0][q]; }
}

// ------------------- K3: per-sample dots via WMMA over a 1024-d chunk
// s1[b][j] = Sum_d z^2*inv[j][d] ; s2[b][j] = Sum_d z*muinv[j][d]
// Two K-steps per iteration with remapped K-slots so every fragment load is a
// contiguous, 16B-aligned b128: half-wave 0 owns d..d+3, half-wave 1 owns d+4..d+7.
template <int DC>
__global__ void k_sample_dots(const float* __restrict__ z,
                              const float* __restrict__ inv,    // [2][D]
                              const float* __restrict__ muinv,  // [2][D]
                              float* __restrict__ partial,      // [nchunk][B][4]
                              int B, int Drt) {
  const int D     = DC ? DC : Drt;
  const int lane  = threadIdx.x & 31;
  const int wave  = threadIdx.x >> 5;
  const int brow  = blockIdx.x * 16;
  const int m     = lane & 15;
  const int khalf = lane >> 4;
  const int dstart = (blockIdx.y * 8 + wave) * 128;   // 128 d per wave

  v8f acc1 = {};   // z^2 x [inv0, inv1, 0...]
  v8f acc2 = {};   // z   x [muinv0, muinv1, 0...]
  const int doff = dstart + 4 * khalf;
  const float* pz = z + (size_t)(brow + m) * D + doff;           // 16B aligned
  // inactive lanes (m>=2) point at row 0 and get masked to zero below
  const float* p1 = inv   + (size_t)(m < 2 ? m : 0) * D + doff;
  const float* p2 = muinv + (size_t)(m < 2 ? m : 0) * D + doff;
  const v4f vzero = { 0.f, 0.f, 0.f, 0.f };
#pragma unroll
  for (int k0 = 0; k0 < 128; k0 += 8) {                          // imm offsets
    v4f az = *(const v4f*)(pz + k0);
    v4f b1 = (m < 2) ? *(const v4f*)(p1 + k0) : vzero;
    v4f b2 = (m < 2) ? *(const v4f*)(p2 + k0) : vzero;
    // WMMA-lo covers elements [0],[1]; WMMA-hi covers [2],[3] of every lane.
    v2f a2lo = { az[0] * az[0], az[1] * az[1] };
    v2f a2hi = { az[2] * az[2], az[3] * az[3] };
    v2f alo  = { az[0], az[1] },  ahi  = { az[2], az[3] };
    v2f b1lo = { b1[0], b1[1] },  b1hi = { b1[2], b1[3] };
    v2f b2lo = { b2[0], b2[1] },  b2hi = { b2[2], b2[3] };
    acc1 = __builtin_amdgcn_wmma_f32_16x16x4_f32(false, a2lo, false, b1lo, (short)0, acc1, false, false);
    acc1 = __builtin_amdgcn_wmma_f32_16x16x4_f32(false, a2hi, false, b1hi, (short)0, acc1, false, false);
    acc2 = __builtin_amdgcn_wmma_f32_16x16x4_f32(false, alo,  false, b2lo, (short)0, acc2, false, false);
    acc2 = __builtin_amdgcn_wmma_f32_16x16x4_f32(false, ahi,  false, b2hi, (short)0, acc2, false, false);
  }
  // Extract C columns N=0,1 (lanes 0,1,16,17 across all 8 acc VGPRs)
  __shared__ float red[8][16][4];   // [wave][row][s1g0 s1g1 s2g0 s2g1]
  if (m < 2) {
    for (int r = 0; r < 8; ++r) {
      const int row = r + 8 * khalf;
      red[wave][row][m]     = acc1[r];
      red[wave][row][2 + m] = acc2[r];
    }
  }
  __syncthreads();
  const int t = threadIdx.x;
  if (t < 64) {
    const int row = t >> 2, st = t & 3;
    float s = 0.f;
    for (int w = 0; w < 8; ++w) s += red[w][row][st];   // fixed order
    partial[((size_t)blockIdx.y * B + (brow + row)) * 4 + st] = s;
  }
}

// --------------------------- K4: chunk reduce, logp, prior, bhatta, outputs
__global__ void k_final(const float* __restrict__ partial, int nchunk,
                        const float* __restrict__ blkpart, int nblk,
                        const float* __restrict__ hdr,
                        const long long* __restrict__ target,
                        float* __restrict__ out, int B, int D) {
  __shared__ float consts[4];     // Sum lsd0, Sum lsd1, c0(=Sum mu^2 inv), c1
  __shared__ float lp[256][2];
  __shared__ int   tg[256];
  __shared__ float sh[256][9];
  const int t = threadIdx.x;

  if (t < 4) {
    float s = 0.f;
    for (int i = 0; i < nblk; ++i) s += blkpart[i * 4 + t];  // fixed order
    consts[t] = s;
  }
  if (t < B) tg[t] = (int)target[t];
  __syncthreads();

  if (t < B) {
    float s10 = 0.f, s11 = 0.f, s20 = 0.f, s21 = 0.f;
    for (int ch = 0; ch < nchunk; ++ch) {                    // fixed order
      const float* p = partial + ((size_t)ch * B + t) * 4;
      s10 += p[0]; s11 += p[1]; s20 += p[2]; s21 += p[3];
    }
    const float base = -0.5f * LOG2PI * (float)D;
    lp[t][0] = base - consts[0] - 0.5f * (s10 - 2.f * s20 + consts[2]);
    lp[t][1] = base - consts[1] - 0.5f * (s11 - 2.f * s21 + consts[3]);
  }
  __syncthreads();

  // per-thread partials: [0]=prior g0, [1]=prior g1,
  // [2]=simnum0 [3]=simcnt0 [4]=simnum1 [5]=simcnt1 [6]=diffnum0 [7]=diffnum1 [8]=diffcnt
  float v[9];
  for (int q = 0; q < 9; ++q) v[q] = 0.f;
  if (t < B) {
    v[tg[t]] = lp[t][tg[t]];
    const int i = t;
    for (int k = 0; k < i; ++k) {                 // tril: row i > col k
      const bool diff = (tg[i] != tg[k]);
      for (int j = 0; j < 2; ++j) {
        const float pw = expf(0.5f * (lp[i][j] + lp[k][j]));
        if (tg[i] == j && tg[k] == j) { v[2 + 2 * j] += pw; v[3 + 2 * j] += 1.f; }
        if (diff) v[6 + j] += pw;
      }
      if (diff) v[8] += 1.f;
    }
  }
  for (int q = 0; q < 9; ++q) sh[t][q] = v[q];
  __syncthreads();
  for (int off = 128; off; off >>= 1) {
    if (t < off) { for (int q = 0; q < 9; ++q) sh[t][q] += sh[t + off][q]; }
    __syncthreads();
  }
  if (t == 0) {
    const float c0 = hdr[0], c1 = hdr[1];
    const float prior = 0.5f * ((sh[0][0] / c0 + hdr[2] / c0) +
                                (sh[0][1] / c1 + hdr[3] / c1));
    const float bc_sim0  = sh[0][2] / sh[0][3];
    const float bc_sim1  = sh[0][4] / sh[0][5];
    const float bc_diff0 = sh[0][6] / sh[0][8];
    const float bc_diff1 = sh[0][7] / sh[0][8];
    const float bhatta = 0.5f * ((1.f - bc_sim0) + bc_diff0 +
                                 (1.f - bc_sim1) + bc_diff1);
    out[0] = prior;
    out[1] = bhatta;
  }
}

// ------------------------------------------------------------------ launcher
extern "C" void kernel_launch(void* const* d_in, const int* in_sizes, int n_in,
                              void* d_out, int out_size, void* d_ws, size_t ws_size,
                              hipStream_t stream) {
  const float*     z      = (const float*)d_in[0];
  const float*     mean   = (const float*)d_in[1];
  const float*     lsd    = (const float*)d_in[2];
  const float*     logdet = (const float*)d_in[3];
  const long long* target = (const long long*)d_in[4];

  const int B = in_sizes[3];                 // 256
  const int D = in_sizes[0] / B;             // 49152
  constexpr int DFIX = 48 * 32 * 32;         // 49152

  float* ws = (float*)d_ws;
  float* sum_mean = ws;  ws += 2 * (size_t)D;
  float* sum_lsd  = ws;  ws += 2 * (size_t)D;
  float* invw     = ws;  ws += 2 * (size_t)D;
  float* muinvw   = ws;  ws += 2 * (size_t)D;
  const int nchunk = D / 1024;               // 48 (8 waves x 128 d)
  float* partial  = ws;  ws += (size_t)nchunk * B * 4;
  const int nblk = D / 256;                  // 192
  float* blkpart  = ws;  ws += (size_t)nblk * 4;
  float* hdr      = ws;  ws += 4;
  unsigned* tbits = (unsigned*)ws;           // B/32 words

  k_counts<<<1, 64, 0, stream>>>(target, logdet, hdr, tbits, B);
  if (D == DFIX) {
    k_group_sums<DFIX><<<D / 128, 256, 0, stream>>>(mean, lsd, tbits, sum_mean, sum_lsd, B, D);
  } else {
    k_group_sums<0><<<D / 128, 256, 0, stream>>>(mean, lsd, tbits, sum_mean, sum_lsd, B, D);
  }
  k_finalize<<<D / 256, 256, 0, stream>>>(sum_mean, sum_lsd, hdr, invw, muinvw, blkpart, D);
  if (D == DFIX) {
    k_sample_dots<DFIX><<<dim3(B / 16, nchunk), 256, 0, stream>>>(z, invw, muinvw, partial, B, D);
  } else {
    k_sample_dots<0><<<dim3(B / 16, nchunk), 256, 0, stream>>>(z, invw, muinvw, partial, B, D);
  }
  k_final<<<1, 256, 0, stream>>>(partial, nchunk, blkpart, nblk, hdr, target, (float*)d_out, B, D);
}